// OrthogonalBivectorBlock_429496730087
// MI455X (gfx1250) — compile-verified
//
#include <hip/hip_runtime.h>
#include <hip/hip_bf16.h>
#include <math.h>

#define PI_F 3.14159265358979323846f

typedef float v2f __attribute__((ext_vector_type(2)));
typedef float v8f __attribute__((ext_vector_type(8)));
typedef int   v4i __attribute__((ext_vector_type(4)));

#define GAS __attribute__((address_space(1)))
#define LAS __attribute__((address_space(3)))

#if defined(__has_builtin)
#if __has_builtin(__builtin_amdgcn_global_load_async_to_lds_b128)
#define HAVE_ASYNC_LDS 1
#endif
#endif

// Problem constants (fixed by the reference)
static constexpr int B_  = 2;
static constexpr int L_  = 1024;
static constexpr int D_  = 512;
static constexpr int M_  = B_ * L_;      // 2048 tokens
static constexpr int CH  = 128;          // scan chunk length
static constexpr int NCH = L_ / CH;      // 8 chunks per batch

// ---------------------------------------------------------------------------
// fp32 WMMA helper: D(16x16) += A(16x4) x B(4x16), all f32.
// Lane layout (wave32): r = lane%16, kh = lane/16.
//   A[j] = Atile[m=r][k = kh*2 + j]
//   B[j] = Btile[k = kh*2 + j][n=r]
//   C/D elem e: m = e + 8*kh, n = r
// ---------------------------------------------------------------------------
__device__ __forceinline__ v8f wmma_f32(v2f a, v2f b, v8f c) {
  return __builtin_amdgcn_wmma_f32_16x16x4_f32(false, a, false, b, (short)0, c,
                                               false, false);
}

// ---------------------------------------------------------------------------
// Generic GEMM  Y = epilogue(X @ W^T + bias)
//   X: (M,K) row-major f32, W: (N,K) row-major f32.
// Each wave computes a 16(M) x 64(N) tile: one A fragment reused over 4 wmmas.
// MODE 0: plain (+bias)   MODE 1: exact GeLU   MODE 2: +bias +residual
// ---------------------------------------------------------------------------
template <int MODE>
__global__ void gemm_wmma_kernel(const float* __restrict__ X,
                                 const float* __restrict__ W,
                                 const float* __restrict__ bias,
                                 const float* __restrict__ resid,
                                 float* __restrict__ out,
                                 int M, int N, int K) {
  const int lane = threadIdx.x & 31;
  const int wid  = blockIdx.x * (blockDim.x >> 5) + (threadIdx.x >> 5);
  const int nT64 = N >> 6;
  const int mt = wid / nT64;
  const int ng = wid % nT64;
  if (mt >= (M >> 4)) return;
  const int m0 = mt << 4;
  const int n0 = ng << 6;
  const int r  = lane & 15;
  const int kh = lane >> 4;

  v8f acc[4] = {};
  const float* xrow = X + (size_t)(m0 + r) * K;
  for (int k0 = 0; k0 < K; k0 += 4) {
    v2f a = *(const v2f*)(xrow + k0 + kh * 2);
#pragma unroll
    for (int t = 0; t < 4; ++t) {
      v2f bfr = *(const v2f*)(W + (size_t)(n0 + t * 16 + r) * K + k0 + kh * 2);
      acc[t] = wmma_f32(a, bfr, acc[t]);
    }
  }
#pragma unroll
  for (int t = 0; t < 4; ++t) {
    const int n  = n0 + t * 16 + r;
    const float bn = bias[n];
#pragma unroll
    for (int e = 0; e < 8; ++e) {
      const int m = m0 + e + 8 * kh;
      float y = acc[t][e] + bn;
      if (MODE == 1) y = 0.5f * y * (1.0f + erff(y * 0.70710678118654752f));
      if (MODE == 2) y += resid[(size_t)m * N + n];
      out[(size_t)m * N + n] = y;
    }
  }
}

// ---------------------------------------------------------------------------
// Key/query phase GEMM (N=16 each): pre = x@W.T + b; t = tanh(pre)*PI;
// writes cos(t), sin(t). One wave per 16x16 tile; ntile 0 -> K, 1 -> Q.
// ---------------------------------------------------------------------------
__global__ void phase_gemm_kernel(const float* __restrict__ x,
                                  const float* __restrict__ Wk,
                                  const float* __restrict__ bk,
                                  const float* __restrict__ Wq,
                                  const float* __restrict__ bq,
                                  float* __restrict__ kc, float* __restrict__ ks,
                                  float* __restrict__ qc, float* __restrict__ qs) {
  const int lane = threadIdx.x & 31;
  const int wid  = blockIdx.x * (blockDim.x >> 5) + (threadIdx.x >> 5);
  const int mt = wid >> 1;          // 128 M tiles
  const int sel = wid & 1;          // 0 = K weights, 1 = Q weights
  if (mt >= (M_ >> 4)) return;
  const int m0 = mt << 4;
  const int r  = lane & 15;
  const int kh = lane >> 4;
  const float* W = sel ? Wq : Wk;
  const float* bb = sel ? bq : bk;
  float* oc = sel ? qc : kc;
  float* os = sel ? qs : ks;

  v8f acc = {};
  const float* xrow = x + (size_t)(m0 + r) * D_;
  const float* wrow = W + (size_t)r * D_;
  for (int k0 = 0; k0 < D_; k0 += 4) {
    v2f a = *(const v2f*)(xrow + k0 + kh * 2);
    v2f b = *(const v2f*)(wrow + k0 + kh * 2);
    acc = wmma_f32(a, b, acc);
  }
  const float bn = bb[r];
#pragma unroll
  for (int e = 0; e < 8; ++e) {
    const int m = m0 + e + 8 * kh;
    float t = tanhf(acc[e] + bn) * PI_F;
    oc[m * 16 + r] = cosf(t);
    os[m * 16 + r] = sinf(t);
  }
}

// ---------------------------------------------------------------------------
// cos/sin of pos_phases[:L] (L*16 elements)
// ---------------------------------------------------------------------------
__global__ void pos_phase_kernel(const float* __restrict__ ph,
                                 float* __restrict__ pc, float* __restrict__ ps) {
  int i = blockIdx.x * blockDim.x + threadIdx.x;
  if (i < L_ * 16) {
    float v = ph[i];
    pc[i] = cosf(v);
    ps[i] = sinf(v);
  }
}

// ---------------------------------------------------------------------------
// Gate logit: g = sigmoid(H @ Wg2.T + bg2), N=1, K=256. One wave per row.
// ---------------------------------------------------------------------------
__global__ void gate2_kernel(const float* __restrict__ H,
                             const float* __restrict__ Wg2,
                             const float* __restrict__ bg2,
                             float* __restrict__ g) {
  const int lane = threadIdx.x & 31;
  const int row  = blockIdx.x * (blockDim.x >> 5) + (threadIdx.x >> 5);
  if (row >= M_) return;
  float s = 0.f;
#pragma unroll
  for (int j = 0; j < 8; ++j) {
    int k = lane + j * 32;
    s = fmaf(H[(size_t)row * 256 + k], Wg2[k], s);
  }
#pragma unroll
  for (int off = 16; off > 0; off >>= 1) s += __shfl_xor(s, off, 32);
  if (lane == 0) g[row] = 1.0f / (1.0f + expf(-(s + bg2[0])));
}

// ---------------------------------------------------------------------------
// S1: per-chunk state totals via fp32 WMMA.
//   sum_{l in chunk} phase[l,sp] * V[l,d]  ==  (16xCH) @ (CHxD) GEMM.
// 4 accumulators (k-real, k-imag, pos-real, pos-imag) share B-fragment (=V).
// One wave per (b, chunk, 16-wide d tile).  cs layout:
//   cs[(((b*4+comp)*16+sp)*NCH + c)*D + d]
// ---------------------------------------------------------------------------
__global__ void chunk_sum_kernel(const float* __restrict__ Vr,
                                 const float* __restrict__ kc,
                                 const float* __restrict__ ks,
                                 const float* __restrict__ pc,
                                 const float* __restrict__ ps,
                                 float* __restrict__ cs) {
  const int lane = threadIdx.x & 31;
  const int wid  = blockIdx.x * (blockDim.x >> 5) + (threadIdx.x >> 5);
  const int dt = wid & 31;            // 32 d-tiles of 16
  const int c  = (wid >> 5) & (NCH - 1);
  const int b  = wid >> 8;
  if (b >= B_) return;
  const int r  = lane & 15;
  const int kh = lane >> 4;
  const int rowV = b * L_ + c * CH;   // also row base for kc/ks
  const int rowP = c * CH;
  const int d0 = dt << 4;

  v8f acc[4] = {};
  for (int k0 = 0; k0 < CH; k0 += 4) {
    const int kk = k0 + kh * 2;
    v2f bf;
    bf.x = Vr[(size_t)(rowV + kk) * D_ + d0 + r];
    bf.y = Vr[(size_t)(rowV + kk + 1) * D_ + d0 + r];
    v2f a0, a1, a2, a3;
    a0.x = kc[(rowV + kk) * 16 + r];  a0.y = kc[(rowV + kk + 1) * 16 + r];
    a1.x = ks[(rowV + kk) * 16 + r];  a1.y = ks[(rowV + kk + 1) * 16 + r];
    a2.x = pc[(rowP + kk) * 16 + r];  a2.y = pc[(rowP + kk + 1) * 16 + r];
    a3.x = ps[(rowP + kk) * 16 + r];  a3.y = ps[(rowP + kk + 1) * 16 + r];
    acc[0] = wmma_f32(a0, bf, acc[0]);
    acc[1] = wmma_f32(a1, bf, acc[1]);
    acc[2] = wmma_f32(a2, bf, acc[2]);
    acc[3] = wmma_f32(a3, bf, acc[3]);
  }
#pragma unroll
  for (int comp = 0; comp < 4; ++comp) {
#pragma unroll
    for (int e = 0; e < 8; ++e) {
      const int sp = e + 8 * kh;
      const int d  = d0 + r;
      cs[((size_t)((b * 4 + comp) * 16 + sp) * NCH + c) * D_ + d] = acc[comp][e];
    }
  }
}

// ---------------------------------------------------------------------------
// S2: in-place exclusive prefix over the NCH chunks. Thread per (b,comp,sp,d).
// ---------------------------------------------------------------------------
__global__ void chunk_scan_kernel(float* __restrict__ cs) {
  int idx = blockIdx.x * blockDim.x + threadIdx.x;
  if (idx >= B_ * 4 * 16 * D_) return;
  const int d    = idx & (D_ - 1);
  const int sp   = (idx >> 9) & 15;
  const int comp = (idx >> 13) & 3;
  const int b    = idx >> 15;
  float* base = cs + ((size_t)((b * 4 + comp) * 16 + sp) * NCH) * D_ + d;
  float run = 0.f;
#pragma unroll
  for (int c = 0; c < NCH; ++c) {
    float v = base[(size_t)c * D_];
    base[(size_t)c * D_] = run;
    run += v;
  }
}

// ---------------------------------------------------------------------------
// S3: chunk-local replay with carry-in + retrieval + gate + norm -> t.
// 64 blocks (b x chunk x 4 d-groups) of 128 threads; thread owns one d.
// Chunk phases (48 KB) staged into LDS via CDNA5 async Global->LDS transfers
// (ASYNCcnt) when available, then broadcast-read in the scan loop.
// ---------------------------------------------------------------------------
__global__ void scan_retrieve_kernel(const float* __restrict__ Vr,
                                     const float* __restrict__ kc,
                                     const float* __restrict__ ks,
                                     const float* __restrict__ qc,
                                     const float* __restrict__ qs,
                                     const float* __restrict__ pc,
                                     const float* __restrict__ ps,
                                     const float* __restrict__ carry,
                                     const float* __restrict__ g,
                                     const float* __restrict__ set_w,
                                     const float* __restrict__ pos_weight,
                                     float* __restrict__ tOut) {
  __shared__ float lds[6 * CH * 16];    // kc,ks,qc,qs,pc,ps for this chunk
  const int tid = threadIdx.x;          // 128
  const int dg = blockIdx.x & 3;
  const int c  = (blockIdx.x >> 2) & (NCH - 1);
  const int b  = blockIdx.x >> 5;
  const int d  = dg * 128 + tid;
  const int row0 = b * L_ + c * CH;

  {
    const float* srcs[6] = {kc + (size_t)row0 * 16, ks + (size_t)row0 * 16,
                            qc + (size_t)row0 * 16, qs + (size_t)row0 * 16,
                            pc + (size_t)(c * CH) * 16, ps + (size_t)(c * CH) * 16};
#ifdef HAVE_ASYNC_LDS
    // CDNA5 async memory->LDS DMA: 16B per lane, no VGPR roundtrip.
    LAS float* ldsw = (LAS float*)&lds[0];
#pragma unroll
    for (int a = 0; a < 6; ++a) {
      // addrspacecast to global AS, then repoint as int4 (drops const).
      GAS const float* sg = (GAS const float*)srcs[a];
#pragma unroll
      for (int i = 0; i < CH * 16; i += 128 * 4) {
        int e = i + tid * 4;
        __builtin_amdgcn_global_load_async_to_lds_b128(
            (GAS v4i*)(sg + e),
            (LAS v4i*)(ldsw + a * CH * 16 + e),
            0, 0);
      }
    }
#if __has_builtin(__builtin_amdgcn_s_wait_asynccnt)
    __builtin_amdgcn_s_wait_asynccnt(0);
#else
    asm volatile("s_wait_asynccnt 0" ::: "memory");
#endif
#else
#pragma unroll
    for (int a = 0; a < 6; ++a)
      for (int i = tid; i < CH * 16; i += 128) lds[a * CH * 16 + i] = srcs[a][i];
#endif
  }
  __syncthreads();

  // softmax(set_w), sigmoid(pos_weight)
  float s0 = set_w[0], s1 = set_w[1], s2 = set_w[2], s3 = set_w[3];
  float mx = fmaxf(fmaxf(s0, s1), fmaxf(s2, s3));
  float e0 = expf(s0 - mx), e1 = expf(s1 - mx), e2 = expf(s2 - mx), e3 = expf(s3 - mx);
  float inv = 1.0f / (e0 + e1 + e2 + e3);
  float w[4] = {e0 * inv, e1 * inv, e2 * inv, e3 * inv};
  float pw = 1.0f / (1.0f + expf(-pos_weight[0]));

  float kr[16], ki[16], pr[16], pi[16];
#pragma unroll
  for (int sp = 0; sp < 16; ++sp) {
    kr[sp] = carry[((size_t)((b * 4 + 0) * 16 + sp) * NCH + c) * D_ + d];
    ki[sp] = carry[((size_t)((b * 4 + 1) * 16 + sp) * NCH + c) * D_ + d];
    pr[sp] = carry[((size_t)((b * 4 + 2) * 16 + sp) * NCH + c) * D_ + d];
    pi[sp] = carry[((size_t)((b * 4 + 3) * 16 + sp) * NCH + c) * D_ + d];
  }

  for (int l = 0; l < CH; ++l) {
    float v = Vr[(size_t)(row0 + l) * D_ + d];
    if (l + 8 < CH)
      __builtin_prefetch(&Vr[(size_t)(row0 + l + 8) * D_ + d], 0, 1);
    const float* KC = &lds[0 * CH * 16 + l * 16];
    const float* KS = &lds[1 * CH * 16 + l * 16];
    const float* QC = &lds[2 * CH * 16 + l * 16];
    const float* QS = &lds[3 * CH * 16 + l * 16];
    const float* PC = &lds[4 * CH * 16 + l * 16];
    const float* PS = &lds[5 * CH * 16 + l * 16];
    float content = 0.f, posv = 0.f;
#pragma unroll
    for (int sp = 0; sp < 16; ++sp) {
      kr[sp] = fmaf(KC[sp], v, kr[sp]);
      ki[sp] = fmaf(KS[sp], v, ki[sp]);
      // Re(mem * conj(q)) = Mr*cos(qp) + Mi*sin(qp)
      float rv = fmaf(kr[sp], QC[sp], ki[sp] * QS[sp]);
      content = fmaf(w[sp >> 2], rv, content);
      pr[sp] = fmaf(PC[sp], v, pr[sp]);
      pi[sp] = fmaf(PS[sp], v, pi[sp]);
      posv = fmaf(pr[sp], PC[sp], fmaf(pi[sp], PS[sp], posv));
    }
    float gg = g[row0 + l];
    float total = gg * content + (1.0f - gg) * pw * posv;
    float pos1 = (float)(c * CH + l + 1);             // 1-based position
    tOut[(size_t)(row0 + l) * D_ + d] = total * rsqrtf(4.0f * pos1);
  }
}

// ---------------------------------------------------------------------------
// LayerNorm over D, in place. One wave (32 lanes x 16 elems) per row.
// ---------------------------------------------------------------------------
__global__ void ln_kernel(float* __restrict__ t, const float* __restrict__ gam,
                          const float* __restrict__ bet) {
  const int lane = threadIdx.x & 31;
  const int row  = blockIdx.x * (blockDim.x >> 5) + (threadIdx.x >> 5);
  if (row >= M_) return;
  float v[16];
  float s = 0.f;
#pragma unroll
  for (int e = 0; e < 16; ++e) {
    v[e] = t[(size_t)row * D_ + lane + e * 32];
    s += v[e];
  }
#pragma unroll
  for (int off = 16; off > 0; off >>= 1) s += __shfl_xor(s, off, 32);
  float mu = s * (1.0f / D_);
  float vs = 0.f;
#pragma unroll
  for (int e = 0; e < 16; ++e) {
    float dv = v[e] - mu;
    vs = fmaf(dv, dv, vs);
  }
#pragma unroll
  for (int off = 16; off > 0; off >>= 1) vs += __shfl_xor(vs, off, 32);
  float rstd = rsqrtf(vs * (1.0f / D_) + 1e-5f);
#pragma unroll
  for (int e = 0; e < 16; ++e) {
    int n = lane + e * 32;
    t[(size_t)row * D_ + n] = (v[e] - mu) * rstd * gam[n] + bet[n];
  }
}

// ---------------------------------------------------------------------------
extern "C" void kernel_launch(void* const* d_in, const int* in_sizes, int n_in,
                              void* d_out, int out_size, void* d_ws, size_t ws_size,
                              hipStream_t stream) {
  const float* x   = (const float*)d_in[0];
  const float* Wk  = (const float*)d_in[1];
  const float* bk  = (const float*)d_in[2];
  const float* Wq  = (const float*)d_in[3];
  const float* bq  = (const float*)d_in[4];
  const float* Wv  = (const float*)d_in[5];
  const float* bv  = (const float*)d_in[6];
  const float* lng = (const float*)d_in[7];
  const float* lnb = (const float*)d_in[8];
  const float* Wo  = (const float*)d_in[9];
  const float* bo  = (const float*)d_in[10];
  const float* sw  = (const float*)d_in[11];
  const float* pph = (const float*)d_in[12];
  const float* pwt = (const float*)d_in[13];
  const float* Wg1 = (const float*)d_in[14];
  const float* bg1 = (const float*)d_in[15];
  const float* Wg2 = (const float*)d_in[16];
  const float* bg2 = (const float*)d_in[17];
  float* out = (float*)d_out;

  // Workspace layout (floats)
  float* wsf = (float*)d_ws;
  float* Vr  = wsf;                      size_t o = (size_t)M_ * D_;     // 1048576
  float* kc  = wsf + o; o += M_ * 16;
  float* ks  = wsf + o; o += M_ * 16;
  float* qc  = wsf + o; o += M_ * 16;
  float* qs  = wsf + o; o += M_ * 16;
  float* pcB = wsf + o; o += L_ * 16;
  float* psB = wsf + o; o += L_ * 16;
  float* H   = wsf + o; o += (size_t)M_ * 256;
  float* g   = wsf + o; o += M_;
  float* cs  = wsf + o; o += (size_t)B_ * 4 * 16 * NCH * D_;
  float* t   = wsf + o; o += (size_t)M_ * D_;
  (void)ws_size; (void)in_sizes; (void)n_in; (void)out_size;

  // 1. key/query phases (WMMA GEMM + tanh/cos/sin epilogue): 256 waves
  phase_gemm_kernel<<<64, 128, 0, stream>>>(x, Wk, bk, Wq, bq, kc, ks, qc, qs);
  // 2. positional cos/sin
  pos_phase_kernel<<<(L_ * 16 + 255) / 256, 256, 0, stream>>>(pph, pcB, psB);
  // 3. Vr = x @ Wv.T + bv (WMMA): 1024 waves
  gemm_wmma_kernel<0><<<256, 128, 0, stream>>>(x, Wv, bv, nullptr, Vr, M_, D_, D_);
  // 4. H = gelu(x @ Wg1.T + bg1) (WMMA): 512 waves
  gemm_wmma_kernel<1><<<128, 128, 0, stream>>>(x, Wg1, bg1, nullptr, H, M_, 256, D_);
  // 5. g = sigmoid(H @ Wg2.T + bg2): wave per row
  gate2_kernel<<<512, 128, 0, stream>>>(H, Wg2, bg2, g);
  // 6. chunk state totals via WMMA: 512 waves
  chunk_sum_kernel<<<128, 128, 0, stream>>>(Vr, kc, ks, pcB, psB, cs);
  // 7. exclusive prefix over chunks (in place)
  chunk_scan_kernel<<<(B_ * 4 * 16 * D_) / 256, 256, 0, stream>>>(cs);
  // 8. chunk-local replay + retrieval + gate + position norm -> t
  scan_retrieve_kernel<<<B_ * NCH * 4, 128, 0, stream>>>(Vr, kc, ks, qc, qs, pcB,
                                                         psB, cs, g, sw, pwt, t);
  // 9. LayerNorm in place
  ln_kernel<<<512, 128, 0, stream>>>(t, lng, lnb);
  // 10. out = x + t @ Wo.T + bo (WMMA + residual): 1024 waves
  gemm_wmma_kernel<2><<<256, 128, 0, stream>>>(t, Wo, bo, x, out, M_, D_, D_);
}